// NormEMAVectorQuantizer_35562329211342
// MI455X (gfx1250) — compile-verified
//
#include <hip/hip_runtime.h>
#include <math.h>

// ---------------------------------------------------------------------------
// NormEMAVectorQuantizer for MI455X (gfx1250, wave32, WMMA + async LDS)
//   z:            [16,1024,128] f32   -> N = 16384 rows, D = 128
//   embedding:    [8192,128]    f32   -> NE = 8192 codes
//   cluster_size: [8192]        f32
//   embed_avg:    [8192,128]    f32
// Outputs (flat, in order): z_q[2097152], loss[1], indices[16384],
//   new_embedding[1048576], new_cluster_size[8192], new_embed_avg[1048576]
// ---------------------------------------------------------------------------

#define N_ROWS   16384
#define N_EMBED  8192
#define DIM      128
#define DECAY    0.99f
#define BETA     1.0f
#define EPSV     1e-5f

typedef __attribute__((ext_vector_type(2))) float v2f;
typedef __attribute__((ext_vector_type(8))) float v8f;
typedef int vq_v4i __attribute__((vector_size(16)));   // matches builtin param

// ---- CDNA5 async global->LDS copy (ASYNCcnt path), with safe fallback -----
#if __has_builtin(__builtin_amdgcn_global_load_async_to_lds_b128) && \
    __has_builtin(__builtin_amdgcn_s_wait_asynccnt)
#define VQ_HAVE_ASYNC 1
__device__ __forceinline__ void vq_async_copy_b128(float* lds_dst, const float* g_src) {
  __builtin_amdgcn_global_load_async_to_lds_b128(
      (__attribute__((address_space(1))) vq_v4i*)g_src,
      (__attribute__((address_space(3))) vq_v4i*)lds_dst, 0, 0);
}
#define VQ_ASYNC_WAIT(n) __builtin_amdgcn_s_wait_asynccnt(n)
#else
#define VQ_HAVE_ASYNC 0
__device__ __forceinline__ void vq_async_copy_b128(float* lds_dst, const float* g_src) {
  *(float4*)lds_dst = *(const float4*)g_src;   // sync fallback
}
#define VQ_ASYNC_WAIT(n) ((void)0)
#endif

// ---- generic block sum (NWAVES waves of 32) -------------------------------
template <int NWAVES>
__device__ inline float block_sum(float v) {
#pragma unroll
  for (int m = 16; m >= 1; m >>= 1) v += __shfl_xor(v, m, 32);
  __shared__ float sm[NWAVES];
  const int w = threadIdx.x >> 5;
  if ((threadIdx.x & 31) == 0) sm[w] = v;
  __syncthreads();
  float r = 0.f;
#pragma unroll
  for (int i = 0; i < NWAVES; ++i) r += sm[i];
  __syncthreads();
  return r;
}

// ---- zero a float region ---------------------------------------------------
__global__ void vq_zero_kernel(float* __restrict__ p, int n) {
  int i = blockIdx.x * 256 + threadIdx.x;
  if (i < n) p[i] = 0.f;
}

// ---- kernel 1: row L2-normalize z -> zf ------------------------------------
__global__ __launch_bounds__(128) void vq_rownorm_kernel(const float* __restrict__ in,
                                                         float* __restrict__ out) {
  const int row = blockIdx.x, t = threadIdx.x;
  float v = in[row * DIM + t];
  float ss = block_sum<4>(v * v);
  float inv = 1.f / fmaxf(sqrtf(ss), 1e-12f);
  out[row * DIM + t] = v * inv;
}

// ---- kernel 2: per-code squared norm --------------------------------------
__global__ __launch_bounds__(128) void vq_enorm_kernel(const float* __restrict__ emb,
                                                       float* __restrict__ enorm2) {
  const int row = blockIdx.x, t = threadIdx.x;
  float v = emb[row * DIM + t];
  float ss = block_sum<4>(v * v);
  if (t == 0) enorm2[row] = ss;
}

// ---- kernel 3: fused WMMA GEMM (zf @ emb^T) + row argmin -------------------
// 4 waves / block, one 16-row tile per wave (64 rows per block).
// B tiles (16x128 f32) are double-buffered in LDS via async global->LDS
// copies; each wave runs 32 x V_WMMA_F32_16X16X4_F32 per column tile.
// B fragments are software-pipelined through registers in groups of 8 so the
// ds_load latency is hidden behind the previous group's 8 WMMAs.
#define B_STRIDE 132                    // floats; 32 lanes hit all 64 banks
#define B_TILE   (16 * B_STRIDE)

__global__ __launch_bounds__(128) void vq_argmin_wmma_kernel(
    const float* __restrict__ zf, const float* __restrict__ emb,
    const float* __restrict__ enorm2, int* __restrict__ idx_out,
    float* __restrict__ idxf_out) {
  __shared__ float smem[2 * B_TILE];    // 16896 B double buffer

  const int tid  = threadIdx.x;
  const int wave = tid >> 5;            // 0..3
  const int lane = tid & 31;
  const int half = lane >> 4;           // K-group select
  const int nrel = lane & 15;           // row (A) / col (B) within tile
  const int rowTile = blockIdx.x * 4 + wave;

  // ---- load A slice into registers: zf row, K = 4s + 2*half + {0,1} ------
  const float* __restrict__ arow = zf + (size_t)(rowTile * 16 + nrel) * DIM;
  v2f a[32];
#pragma unroll
  for (int s = 0; s < 32; ++s) a[s] = *(const v2f*)(arow + 4 * s + 2 * half);

  float best[8];
  int bidx[8];
#pragma unroll
  for (int r = 0; r < 8; ++r) { best[r] = 3.0e38f; bidx[r] = 0; }

  // ---- async fill of one 16x128 B tile (8 KB) = 16 lane-wide b128 ops -----
  // chunk c = wave*4 + j; piece p = c*32 + lane covers 16 bytes.
  auto fill = [&](int buf, int ct) {
#pragma unroll
    for (int j = 0; j < 4; ++j) {
      const int p = (wave * 4 + j) * 32 + lane;  // 0..511
      const int row = p >> 5, off16 = p & 31;    // row 0..15, 16B piece 0..31
      const float* g = emb + (size_t)(ct * 16 + row) * DIM + off16 * 4;
      float* l = smem + buf * B_TILE + row * B_STRIDE + off16 * 4;
      vq_async_copy_b128(l, g);
    }
  };

  fill(0, 0);

  for (int ct = 0; ct < N_EMBED / 16; ++ct) {
    const int cur = ct & 1;
    if (ct + 1 < N_EMBED / 16) {
      fill(cur ^ 1, ct + 1);   // 4 async ops/wave for the next buffer
      VQ_ASYNC_WAIT(4);        // previous 4 (this buffer) are complete
    } else {
      VQ_ASYNC_WAIT(0);
    }
    __syncthreads();           // tile visible to all waves

    const float* __restrict__ bbase = smem + cur * B_TILE + nrel * B_STRIDE;
    v8f acc = {};

    // software-pipelined B fragments: groups of 8 K-steps, double-buffered
    v2f bf[2][8];
#pragma unroll
    for (int j = 0; j < 8; ++j)
      bf[0][j] = *(const v2f*)(bbase + 4 * j + 2 * half);
#pragma unroll
    for (int g = 0; g < 4; ++g) {
      if (g < 3) {
#pragma unroll
        for (int j = 0; j < 8; ++j)
          bf[(g + 1) & 1][j] = *(const v2f*)(bbase + 4 * (8 * (g + 1) + j) + 2 * half);
      }
#pragma unroll
      for (int j = 0; j < 8; ++j) {
        acc = __builtin_amdgcn_wmma_f32_16x16x4_f32(
            /*neg_a=*/false, a[8 * g + j], /*neg_b=*/false, bf[g & 1][j],
            /*c_mod=*/(short)0, acc, /*reuse_a=*/false, /*reuse_b=*/false);
      }
    }

    // distance (up to per-row constant): ||e_n||^2 - 2*dot
    const float en = enorm2[ct * 16 + nrel];
    const int col = ct * 16 + nrel;
#pragma unroll
    for (int r = 0; r < 8; ++r) {
      float v = fmaf(-2.f, acc[r], en);
      if (v < best[r]) { best[r] = v; bidx[r] = col; }  // earliest col wins ties
    }
    __syncthreads();           // all waves done reading before next overwrite
  }

  // reduce across the 16 lanes of each half (xor masks stay within half)
#pragma unroll
  for (int m = 1; m < 16; m <<= 1) {
#pragma unroll
    for (int r = 0; r < 8; ++r) {
      float ov = __shfl_xor(best[r], m, 32);
      int oi = __shfl_xor(bidx[r], m, 32);
      if (ov < best[r] || (ov == best[r] && oi < bidx[r])) {
        best[r] = ov;
        bidx[r] = oi;
      }
    }
  }

  if (nrel == 0) {  // lane 0 holds rows 0..7, lane 16 holds rows 8..15
#pragma unroll
    for (int r = 0; r < 8; ++r) {
      const int row = rowTile * 16 + 8 * half + r;
      idx_out[row] = bidx[r];
      idxf_out[row] = (float)bidx[r];
    }
  }
}

// ---- kernel 4: gather z_q, per-row loss partial, scatter segment sums -----
__global__ __launch_bounds__(128) void vq_gather_scatter_kernel(
    const int* __restrict__ idx, const float* __restrict__ emb,
    const float* __restrict__ zf, float* __restrict__ zq_out,
    float* __restrict__ embed_sum, float* __restrict__ bins,
    float* __restrict__ loss_rows) {
  const int row = blockIdx.x, t = threadIdx.x;
  const int e = idx[row];
  const float ev = emb[(size_t)e * DIM + t];
  zq_out[(size_t)row * DIM + t] = ev;           // straight-through output == z_q
  const float zv = zf[(size_t)row * DIM + t];
  const float d = ev - zv;
  const float s = block_sum<4>(d * d);
  atomicAdd(&embed_sum[(size_t)e * DIM + t], zv);
  if (t == 0) {
    atomicAdd(&bins[e], 1.0f);                  // +1.0f adds are order-exact
    loss_rows[row] = s;
  }
}

// ---- kernel 5: EMA + renormalize codebook ---------------------------------
__global__ __launch_bounds__(128) void vq_codebook_update_kernel(
    const float* __restrict__ bins, const float* __restrict__ embed_sum,
    const float* __restrict__ emb, const float* __restrict__ cluster_size,
    const float* __restrict__ embed_avg, float* __restrict__ out_embedding,
    float* __restrict__ out_cluster, float* __restrict__ out_avg) {
  const int n = blockIdx.x, t = threadIdx.x;
  const float b = bins[n];
  if (t == 0) out_cluster[n] = cluster_size[n] * DECAY + (1.f - DECAY) * b;

  const float s = embed_sum[(size_t)n * DIM + t] / (b + EPSV);
  const float ss = block_sum<4>(s * s);
  const float inv = 1.f / fmaxf(sqrtf(ss), 1e-12f);
  const float en = (b == 0.f) ? emb[(size_t)n * DIM + t] : s * inv;

  const float avg = embed_avg[(size_t)n * DIM + t] * DECAY + (1.f - DECAY) * en;
  out_avg[(size_t)n * DIM + t] = avg;

  const float ss2 = block_sum<4>(avg * avg);
  const float inv2 = 1.f / fmaxf(sqrtf(ss2), 1e-12f);
  out_embedding[(size_t)n * DIM + t] = avg * inv2;
}

// ---- kernel 6: deterministic loss reduction -------------------------------
__global__ __launch_bounds__(256) void vq_loss_finalize_kernel(
    const float* __restrict__ loss_rows, float* __restrict__ loss_out) {
  float s = 0.f;
  for (int i = threadIdx.x; i < N_ROWS; i += 256) s += loss_rows[i];
  const float total = block_sum<8>(s);
  if (threadIdx.x == 0)
    *loss_out = BETA * total / (float)(N_ROWS * DIM);
}

// ---------------------------------------------------------------------------
extern "C" void kernel_launch(void* const* d_in, const int* in_sizes, int n_in,
                              void* d_out, int out_size, void* d_ws,
                              size_t ws_size, hipStream_t stream) {
  const float* z = (const float*)d_in[0];
  const float* embedding = (const float*)d_in[1];
  const float* cluster_size = (const float*)d_in[2];
  const float* embed_avg = (const float*)d_in[3];

  float* out = (float*)d_out;
  float* zq_out = out;                       // 2097152
  float* loss_out = out + 2097152;           // 1
  float* idxf_out = out + 2097153;           // 16384
  float* newemb_out = out + 2113537;         // 1048576
  float* newcs_out = out + 3162113;          // 8192
  float* newavg_out = out + 3170305;         // 1048576

  float* ws = (float*)d_ws;
  float* zf = ws;                            // 2097152
  float* enorm2 = ws + 2097152;              // 8192
  float* bins = ws + 2105344;                // 8192
  float* embed_sum = ws + 2113536;           // 1048576
  float* loss_rows = ws + 3162112;           // 16384
  int* idx_ws = (int*)(ws + 3178496);        // 16384 ints

  // zero bins + embed_sum (contiguous region)
  const int nz = 8192 + 1048576;
  vq_zero_kernel<<<(nz + 255) / 256, 256, 0, stream>>>(bins, nz);

  vq_rownorm_kernel<<<N_ROWS, 128, 0, stream>>>(z, zf);
  vq_enorm_kernel<<<N_EMBED, 128, 0, stream>>>(embedding, enorm2);

  vq_argmin_wmma_kernel<<<N_ROWS / 64, 128, 0, stream>>>(zf, embedding, enorm2,
                                                         idx_ws, idxf_out);

  vq_gather_scatter_kernel<<<N_ROWS, 128, 0, stream>>>(
      idx_ws, embedding, zf, zq_out, embed_sum, bins, loss_rows);

  vq_codebook_update_kernel<<<N_EMBED, 128, 0, stream>>>(
      bins, embed_sum, embedding, cluster_size, embed_avg, newemb_out,
      newcs_out, newavg_out);

  vq_loss_finalize_kernel<<<1, 256, 0, stream>>>(loss_rows, loss_out);
}